// LinkConv_5755256177464
// MI455X (gfx1250) — compile-verified
//
#include <hip/hip_runtime.h>
#include <math.h>

#define N_NODES 50000
#define N_EDGES 800000
#define H_DIM 128
#define TWO_H 256

typedef __attribute__((ext_vector_type(2))) float v2f;
typedef __attribute__((ext_vector_type(8))) float v8f;

// ---------------- zero scratch ----------------
__global__ void lc_zero_kernel(float* __restrict__ p, int n) {
    int i = blockIdx.x * blockDim.x + threadIdx.x;
    if (i < n) p[i] = 0.0f;
}

// ---------------- edge aggregation: m[dst, col_off + :128] += feat[e] * h[src, :] ----------------
// One (edge, lane) unit handles 4 consecutive channels: float4 gather + 4 float atomics.
__global__ void lc_agg_kernel(const float* __restrict__ h,
                              const float* __restrict__ feat,
                              const int* __restrict__ src,
                              const int* __restrict__ dst,
                              float* __restrict__ m,
                              int col_off) {
    long long gid    = (long long)blockIdx.x * blockDim.x + threadIdx.x;
    long long total  = (long long)N_EDGES * 32;
    long long stride = (long long)gridDim.x * blockDim.x;
    for (; gid < total; gid += stride) {
        int e    = (int)(gid >> 5);
        int lane = (int)(gid & 31);
        float f = feat[e];
        int s = src[e];
        int d = dst[e];
        float4 v = ((const float4*)(h + (size_t)s * H_DIM))[lane];
        float* mb = m + (size_t)d * TWO_H + col_off + lane * 4;
        atomicAdd(mb + 0, f * v.x);
        atomicAdd(mb + 1, f * v.y);
        atomicAdd(mb + 2, f * v.z);
        atomicAdd(mb + 3, f * v.w);
    }
}

// ---------------- GEMM [N,256]x[256,128] + bias + exact GELU + residual ----------------
// One wave per 16x16 output tile using V_WMMA_F32_16X16X4_F32 (full fp32 path).
// Block = 256 threads = 8 waves; wave w covers output cols [16w, 16w+16).
__global__ void lc_gemm_gelu_kernel(const float* __restrict__ m,    // [N, 256]
                                    const float* __restrict__ W,    // [256, 128]
                                    const float* __restrict__ bias, // [128]
                                    const float* h_in,              // [N, 128] (may alias h_out)
                                    float* h_out) {                 // [N, 128]
    const int row0 = blockIdx.x * 16;
    const int wave = threadIdx.x >> 5;
    const int lane = threadIdx.x & 31;
    const int col0 = wave * 16;

    const int mn    = lane & 15;          // M (for A) / N (for B, C/D)
    const int khalf = (lane >> 4) * 2;    // lanes 0-15 -> K base 0; lanes 16-31 -> K base 2

    v8f acc = {};

    const float* arow = m + (size_t)(row0 + mn) * TWO_H + khalf; // A: row = m-row, contiguous K pair
    const float* bcol = W + (size_t)khalf * H_DIM + col0 + mn;   // B: col = W col, K strided by 128

    #pragma unroll 8
    for (int k0 = 0; k0 < TWO_H; k0 += 4) {
        v2f a, b;
        a.x = arow[k0 + 0];
        a.y = arow[k0 + 1];
        b.x = bcol[(size_t)(k0 + 0) * H_DIM];
        b.y = bcol[(size_t)(k0 + 1) * H_DIM];
        // D = A(16x4,f32) * B(4x16,f32) + C(16x16,f32)
        acc = __builtin_amdgcn_wmma_f32_16x16x4_f32(
            /*neg_a=*/false, a, /*neg_b=*/false, b,
            /*c_mod=*/(short)0, acc, /*reuse_a=*/false, /*reuse_b=*/false);
    }

    const float bval  = bias[col0 + mn];
    const int   mbase = (lane >> 4) * 8;  // C/D: lanes 0-15 hold M=v, lanes 16-31 hold M=v+8

    #pragma unroll
    for (int v = 0; v < 8; ++v) {
        const int row = row0 + mbase + v;
        const int col = col0 + mn;
        float x = acc[v] + bval;
        float g = 0.5f * x * (1.0f + erff(x * 0.70710678118654752f)); // exact GELU
        const size_t idx = (size_t)row * H_DIM + col;
        h_out[idx] = h_in[idx] + g;
    }
}

extern "C" void kernel_launch(void* const* d_in, const int* in_sizes, int n_in,
                              void* d_out, int out_size, void* d_ws, size_t ws_size,
                              hipStream_t stream) {
    const float* h       = (const float*)d_in[0];
    const float* feat_ct = (const float*)d_in[1];
    const float* feat_cb = (const float*)d_in[2];
    const int*   src_ct  = (const int*)d_in[3];
    const int*   dst_ct  = (const int*)d_in[4];
    const int*   src_cb  = (const int*)d_in[5];
    const int*   dst_cb  = (const int*)d_in[6];
    const float* W0      = (const float*)d_in[7];
    const float* b0      = (const float*)d_in[8];
    const float* W1      = (const float*)d_in[9];
    const float* b1      = (const float*)d_in[10];

    float* out = (float*)d_out;
    float* m   = (float*)d_ws;   // [N_NODES, 256] fp32 scratch = 51.2 MB

    const int m_elems   = N_NODES * TWO_H;
    const int zero_blks = (m_elems + 255) / 256;
    const int agg_blks  = (int)(((long long)N_EDGES * 32) / 256); // exact: 100000
    const int gemm_blks = N_NODES / 16;                           // exact: 3125

    const float* Ws[2] = {W0, W1};
    const float* bs[2] = {b0, b1};
    const float* hin = h;

    for (int layer = 0; layer < 2; ++layer) {
        lc_zero_kernel<<<zero_blks, 256, 0, stream>>>(m, m_elems);
        lc_agg_kernel<<<agg_blks, 256, 0, stream>>>(hin, feat_ct, src_ct, dst_ct, m, 0);
        lc_agg_kernel<<<agg_blks, 256, 0, stream>>>(hin, feat_cb, src_cb, dst_cb, m, H_DIM);
        lc_gemm_gelu_kernel<<<gemm_blks, 256, 0, stream>>>(m, Ws[layer], bs[layer], hin, out);
        hin = out;
    }
}